// ComplexMult_34909494182082
// MI455X (gfx1250) — compile-verified
//
#include <hip/hip_runtime.h>

// Complex block-diagonal GEMM for MI455X (gfx1250), f32 WMMA 16x16x4.
// out[k,o,m] = sum_i A[k,i,m]*wr[k,i,o] + A3[k,i,m]*wi[k,i,o]
//   with m = interleaved re/im float index (0..2*H*W-1),
//   A3[i,m] = sgn(m) * A[i, m^1],  sgn = -1 for even m, +1 for odd m.

typedef __attribute__((ext_vector_type(2))) float v2f;
typedef __attribute__((ext_vector_type(8))) float v8f;

#define NB     8            // blocks k
#define IDIM   96           // input channels per block
#define ODIM   96           // output channels per block
#define MF     130320       // floats per (k,i) plane = 360*181*2
#define KSTEPS 24           // 96 / 4 (WMMA K = 4)
#define STRIPW 64           // plane-floats per LDS strip (4 subtiles of 16)
#define NSTRIP 2037         // ceil(130320 / 64)
#define PAIRSTRIDE 160      // floats per interleaved row-pair in LDS (128 + 32 pad)

__global__ __launch_bounds__(192)
void ComplexMult_34909494182082_kernel(const float* __restrict__ inp,
                                       const float* __restrict__ wgt,
                                       const float* __restrict__ bias,
                                       float* __restrict__ out) {
    __shared__ float lds[48 * PAIRSTRIDE];   // 30720 B: 96 rows x 64 cols, pair-interleaved

    const int k      = blockIdx.y;
    const int tid    = threadIdx.x;          // 0..191
    const int wave   = tid >> 5;             // 0..5 -> o-tile
    const int lane   = tid & 31;
    const int laneM  = lane & 15;            // M (A) / N (B,C) position
    const int laneHi = lane >> 4;            // 0: K=0,1   1: K=2,3
    const int o0     = wave * 16;
    const float sgn  = (laneM & 1) ? 1.0f : -1.0f;

    const float* ik = inp  + (size_t)k * IDIM * MF;
    const float* wk = wgt  + (size_t)k * IDIM * ODIM * 2;
    const float* bk = bias + (size_t)k * ODIM * 2;
    float*       ok = out  + (size_t)k * ODIM * MF;

    // ---- Weight strip in registers, WMMA A-operand layout (M=o_local, K=i_local) ----
    // lane L (0-15): M=L, VGPR0=K0, VGPR1=K1 ; lane L+16: same M, K2/K3.
    v2f wAr[KSTEPS], wAi[KSTEPS];
#pragma unroll
    for (int s = 0; s < KSTEPS; ++s) {
        const int i0 = 4 * s + 2 * laneHi;
        const float* p = wk + ((size_t)i0 * ODIM + o0 + laneM) * 2;
        const float2 w0 = *(const float2*)p;               // (wr,wi) at row i0
        const float2 w1 = *(const float2*)(p + ODIM * 2);  // (wr,wi) at row i0+1
        wAr[s] = (v2f){w0.x, w1.x};
        wAi[s] = (v2f){w0.y, w1.y};
    }

    // ---- Bias per lane for the 8 C/D VGPRs: M = r + 8*laneHi, c = parity(N) = laneM&1 ----
    float bv[8];
#pragma unroll
    for (int r = 0; r < 8; ++r)
        bv[r] = bk[(size_t)(o0 + r + 8 * laneHi) * 2 + (laneM & 1)];

    for (int strip = blockIdx.x; strip < NSTRIP; strip += gridDim.x) {
        const int m0 = strip * STRIPW;

        __syncthreads();   // previous strip's LDS reads complete

        // ---- Cooperative stage: 96 rows x 64 floats, rows pair-interleaved.
        // lds[p*PAIRSTRIDE + col*2 + e] = A[2p+e][m0+col].  768 pair-chunks / 192 thr = 4 each.
#pragma unroll
        for (int j = 0; j < 4; ++j) {
            const int flat = tid + 192 * j;       // 0..767
            const int p    = flat >> 4;           // row pair 0..47
            const int c4   = flat & 15;           // float4 column 0..15
            const int gm   = m0 + c4 * 4;
            if (gm + 4 <= MF) {
                const float* g = ik + (size_t)(2 * p) * MF + gm;
                const float4 f0 = *(const float4*)g;          // row 2p
                const float4 f1 = *(const float4*)(g + MF);   // row 2p+1
                float* l = lds + p * PAIRSTRIDE + c4 * 8;
                *(float2*)(l + 0) = float2{f0.x, f1.x};
                *(float2*)(l + 2) = float2{f0.y, f1.y};
                *(float2*)(l + 4) = float2{f0.z, f1.z};
                *(float2*)(l + 6) = float2{f0.w, f1.w};
            }
        }

        __syncthreads();

        // ---- 16(o) x 64(m) output tile: 24 K-steps x 4 subtiles x 2 WMMA ----
        const v8f z = {0.f, 0.f, 0.f, 0.f, 0.f, 0.f, 0.f, 0.f};
        v8f acc[4] = {z, z, z, z};

        // Straight and column-swapped per-lane LDS bases (pair p = 2s + laneHi).
        const float* lb  = lds + laneHi * PAIRSTRIDE + laneM * 2;
        const float* lbs = lds + laneHi * PAIRSTRIDE + (laneM ^ 1) * 2;

#pragma unroll
        for (int s = 0; s < KSTEPS; ++s) {
            const float* row  = lb  + s * (2 * PAIRSTRIDE);
            const float* rows = lbs + s * (2 * PAIRSTRIDE);
#pragma unroll
            for (int t = 0; t < 4; ++t) {
                v2f b  = *(const v2f*)(row  + t * 32);   // {A[i0][m], A[i0+1][m]}
                v2f b3 = *(const v2f*)(rows + t * 32);   // {A[i0][m^1], A[i0+1][m^1]}
                b3 = b3 * sgn;
                acc[t] = __builtin_amdgcn_wmma_f32_16x16x4_f32(
                    false, wAr[s], false, b, (short)0, acc[t], false, false);
                acc[t] = __builtin_amdgcn_wmma_f32_16x16x4_f32(
                    false, wAi[s], false, b3, (short)0, acc[t], false, false);
            }
        }

        // ---- Epilogue: bias + coalesced stores (wave-uniform guards keep EXEC full) ----
#pragma unroll
        for (int t = 0; t < 4; ++t) {
            const int ms = m0 + t * 16;
            if (ms < MF) {
                float* op = ok + (size_t)(o0 + 8 * laneHi) * MF + ms + laneM;
#pragma unroll
                for (int r = 0; r < 8; ++r)
                    op[(size_t)r * MF] = acc[t][r] + bv[r];
            }
        }
    }
}

extern "C" void kernel_launch(void* const* d_in, const int* in_sizes, int n_in,
                              void* d_out, int out_size, void* d_ws, size_t ws_size,
                              hipStream_t stream) {
    (void)in_sizes; (void)n_in; (void)out_size; (void)d_ws; (void)ws_size;
    const float* inp  = (const float*)d_in[0];
    const float* wgt  = (const float*)d_in[1];
    const float* bias = (const float*)d_in[2];
    float*       out  = (float*)d_out;

    dim3 grid(256, NB);   // 2048 WGs x 6 waves: ~8 m-strips per WG amortize weight loads
    dim3 block(192);      // 6 waves of 32 = 6 o-tiles covering O=96
    ComplexMult_34909494182082_kernel<<<grid, block, 0, stream>>>(inp, wgt, bias, out);
}